// FsCoding1_20684562497553
// MI455X (gfx1250) — compile-verified
//
#include <hip/hip_runtime.h>

// FS-coding elementwise map on gfx1250 (wave32).
// Streaming kernel; the 16-step recurrence is executed two elements at a time
// via v_pk_fma_f32. CDNA5 repurposes VOP3P OPSEL bits on f32 packed ops
// (reuse hints), so the legal encoding is all-zero op_sel/op_sel_hi — the
// asm parser's gfx90a-style default op_sel_hi:[1,1,1] is rejected, hence the
// fully explicit modifiers below. Per step / 2 elements:
//   pk_fma (v-updates) + 2x(v_cmp+v_cndmask) + pk_fma (out-updates) = 6 ops
// i.e. 3 VALU issue slots per element per step instead of 4.

typedef float v4f __attribute__((ext_vector_type(4)));
typedef float v2f __attribute__((ext_vector_type(2)));

__device__ __forceinline__ unsigned fbits(float f) {
    return __builtin_bit_cast(unsigned, f);
}
__device__ __forceinline__ unsigned long long splat2(float f) {
    unsigned u = fbits(f);
    return (unsigned long long)u | ((unsigned long long)u << 32);
}

// Hardcoded K=16 coefficients from the reference.
#define FS_H { -0.65453446f, 0.4741694f, 1.5345732f, 0.34063083f, \
                0.10334169f, 1.5662742f, 1.5825152f, 0.08264282f, \
                1.271305f,   0.05652184f, 0.8087557f, 0.48658752f, \
               -0.98152035f, 5.504204f,  3.9964867f, -19.865507f }
#define FS_D {  0.47450694f, 0.61293966f, 0.34035066f, -0.06458966f, \
                0.37327227f, 0.26768795f, 0.08609799f,  0.21590698f, \
                0.06298151f, 0.02074145f, 0.07180741f, -0.1188115f,  \
                0.08973021f, 0.06638855f, 0.02828525f,  0.00312297f }
#define FS_T {  0.46731842f, 0.8967325f,  0.32910728f, -1.0000211f, \
                0.18526751f, 1.000173f,   0.01329661f,  0.06418973f, \
               -0.01542763f, 1.0000025f, -0.08678632f,  0.9953476f,  \
                0.70950633f, -0.5301457f, -1.300978f,  -1.0005718f }

// Scalar path (tail elements) — plain C, bit-identical math.
// (v - T)/(|v|+1) > 0  <=>  v > T  since the denominator is strictly positive.
static __device__ __forceinline__ float fs_elem_scalar(float x) {
    const float H[16] = FS_H;
    const float D[16] = FS_D;
    const float T[16] = FS_T;
    float v = __builtin_fabsf(x), z = 0.0f, out = 0.0f;
#pragma unroll
    for (int t = 0; t < 16; ++t) {
        v   = __builtin_fmaf(z, -H[t], v);
        z   = (v > T[t]) ? 1.0f : 0.0f;
        out = __builtin_fmaf(z, D[t], out);
    }
    float s = (x > 0.0f) ? 1.0f : ((x < 0.0f) ? -1.0f : x);
    return out * s;
}

__global__ void __launch_bounds__(256)
fs_coding1_kernel(const float* __restrict__ x, float* __restrict__ y, long long n) {
    const float H[16] = FS_H;
    const float D[16] = FS_D;
    const float T[16] = FS_T;

    const long long n4     = n >> 2;
    const long long tid    = (long long)blockIdx.x * blockDim.x + threadIdx.x;
    const long long stride = (long long)gridDim.x * blockDim.x;

    const v4f* __restrict__ xv = (const v4f*)x;
    v4f* __restrict__ yv       = (v4f*)y;

    for (long long i = tid; i < n4; i += stride) {
        v4f xin = __builtin_nontemporal_load(xv + i);

        // Two element-pairs per float4: {e0,e1} and {e2,e3}.
        v2f vp[2];   // packed v accumulators
        v2f op[2];   // packed out accumulators
        v2f zp[2];   // packed gates (0.0 / 1.0)
#pragma unroll
        for (int p = 0; p < 2; ++p) {
            vp[p].x = __builtin_fabsf(xin[2 * p + 0]);
            vp[p].y = __builtin_fabsf(xin[2 * p + 1]);
            op[p].x = 0.0f;  op[p].y = 0.0f;
            zp[p].x = 0.0f;  zp[p].y = 0.0f;
        }

#pragma unroll
        for (int t = 0; t < 16; ++t) {
            const unsigned long long hh = splat2(-H[t]);  // {-h,-h} in SGPR pair
            const unsigned long long dd = splat2(D[t]);   // { d, d} in SGPR pair
#pragma unroll
            for (int p = 0; p < 2; ++p) {
                // v -= z_prev * h  (both lanes, exact: z in {0,1})
                asm("v_pk_fma_f32 %0, %1, %2, %0 op_sel:[0,0,0] op_sel_hi:[0,0,0]"
                    : "+v"(vp[p]) : "v"(zp[p]), "s"(hh));
                // z = (v > T) ? 1 : 0  (division in reference eliminated:
                // denominator |v|+1 > 0, so only the sign of v-T matters)
                zp[p].x = (vp[p].x > T[t]) ? 1.0f : 0.0f;
                zp[p].y = (vp[p].y > T[t]) ? 1.0f : 0.0f;
                // out += z * d  (both lanes)
                asm("v_pk_fma_f32 %0, %1, %2, %0 op_sel:[0,0,0] op_sel_hi:[0,0,0]"
                    : "+v"(op[p]) : "v"(zp[p]), "s"(dd));
            }
        }

        v4f r;
#pragma unroll
        for (int k = 0; k < 4; ++k) {
            float xk = xin[k];
            // jnp.sign semantics: +/-1, +/-0 passthrough, NaN passthrough.
            float s = (xk > 0.0f) ? 1.0f : ((xk < 0.0f) ? -1.0f : xk);
            float o = (k & 1) ? op[k >> 1].y : op[k >> 1].x;
            r[k] = o * s;
        }
        __builtin_nontemporal_store(r, yv + i);
    }

    // Scalar tail (n % 4); empty for n = 2^27 but kept for generality.
    for (long long j = (n4 << 2) + tid; j < n; j += stride) {
        __builtin_nontemporal_store(fs_elem_scalar(x[j]), y + j);
    }
}

extern "C" void kernel_launch(void* const* d_in, const int* in_sizes, int n_in,
                              void* d_out, int out_size, void* d_ws, size_t ws_size,
                              hipStream_t stream) {
    (void)n_in; (void)out_size; (void)d_ws; (void)ws_size;
    const float* x = (const float*)d_in[0];
    float*       y = (float*)d_out;
    const long long n  = (long long)in_sizes[0];   // 32*2048*2048 = 134,217,728
    const long long n4 = n >> 2;

    const int block = 256;                         // 8 wave32 per workgroup
    long long threads_wanted = (n4 + 7) / 8;       // ~8 float4 packets per thread
    long long grid = (threads_wanted + block - 1) / block;
    if (grid < 1) grid = 1;
    if (grid > 1048576) grid = 1048576;

    fs_coding1_kernel<<<(int)grid, block, 0, stream>>>(x, y, n);
}